// ColorCodeGNN_45028437131604
// MI455X (gfx1250) — compile-verified
//
#include <hip/hip_runtime.h>

#define N_NODES  50000
#define N_EDGES  600000
#define IN_DIM   4
#define HID      128
#define N_LAYERS 6
#define BN_EPS   1e-5f

typedef __attribute__((ext_vector_type(2))) float v2f;
typedef __attribute__((ext_vector_type(8))) float v8f;

__device__ __forceinline__ void atomic_add_f32(float* p, float v) {
    // lowers to global_atomic_add_f32 on gfx1250
    unsafeAtomicAdd(p, v);
}

// ---------------------------------------------------------------------------
// Degree: deg[dst] += 1 over edges (self-loop +1 added in finisher)
// ---------------------------------------------------------------------------
__global__ void degree_kernel(const int* __restrict__ dst, float* __restrict__ deg, int e_cnt) {
    int e = blockIdx.x * blockDim.x + threadIdx.x;
    if (e < e_cnt) atomic_add_f32(&deg[dst[e]], 1.0f);
}

__global__ void inv_sqrt_deg_kernel(const float* __restrict__ deg, float* __restrict__ isd, int n) {
    int i = blockIdx.x * blockDim.x + threadIdx.x;
    if (i < n) isd[i] = rsqrtf(deg[i] + 1.0f);
}

// ---------------------------------------------------------------------------
// Encoder: h = relu(x @ enc_w + enc_b), x:[N,4], enc_w:[4,128]
// thread = (node, 4 channels): float4 stores, x row broadcast within wave
// ---------------------------------------------------------------------------
__global__ void encoder_kernel(const float* __restrict__ x, const float* __restrict__ w,
                               const float* __restrict__ b, float* __restrict__ h, int n) {
    int idx = blockIdx.x * blockDim.x + threadIdx.x;   // over n*32
    if (idx >= n * (HID / 4)) return;
    int i  = idx >> 5;
    int c4 = (idx & 31) * 4;
    float4 acc = *(const float4*)&b[c4];
    float4 xr  = *(const float4*)&x[i * IN_DIM];        // IN_DIM == 4
    const float xv[4] = {xr.x, xr.y, xr.z, xr.w};
#pragma unroll
    for (int k = 0; k < IN_DIM; ++k) {
        float4 wr = *(const float4*)&w[k * HID + c4];
        acc.x += xv[k] * wr.x; acc.y += xv[k] * wr.y;
        acc.z += xv[k] * wr.z; acc.w += xv[k] * wr.w;
    }
    acc.x = fmaxf(acc.x, 0.f); acc.y = fmaxf(acc.y, 0.f);
    acc.z = fmaxf(acc.z, 0.f); acc.w = fmaxf(acc.w, 0.f);
    *(float4*)&h[(long)i * HID + c4] = acc;
}

// ---------------------------------------------------------------------------
// WMMA fp32 GEMM: OUT[N,128] = H[N,128] @ W[128,128]
// block = 256 threads (8 waves). Block tile: 16 rows x 128 cols.
// Wave w computes the 16x16 tile at columns [16w,16w+16) via 32 k-steps of
// V_WMMA_F32_16X16X4_F32.
// A layout (ISA 7.12.2, 32-bit A 16x4): lanes 0-15 -> K=k+{0,1},
// lanes 16-31 -> K=k+{2,3}; M = lane&15.
// C/D layout: VGPR r -> M = r + 8*(lane>=16), N = lane&15.
// ---------------------------------------------------------------------------
__global__ __launch_bounds__(256) void gemm_wmma_f32(const float* __restrict__ H,
                                                     const float* __restrict__ W,
                                                     float* __restrict__ OUT, int n) {
    __shared__ float Atile[16 * HID];   // 8 KB
    const int rowBase = blockIdx.x * 16;
    const int tid = threadIdx.x;

    // cooperative stage of the 16x128 A tile (float4 per thread x2)
    for (int i = tid; i < 16 * (HID / 4); i += 256) {
        int r = i >> 5, c4 = (i & 31) * 4;
        int row = rowBase + r;
        float4 v = make_float4(0.f, 0.f, 0.f, 0.f);
        if (row < n) v = *(const float4*)&H[(long)row * HID + c4];
        *(float4*)&Atile[r * HID + c4] = v;
    }
    __syncthreads();

    const int wave    = tid >> 5;          // 0..7 -> 16-wide column tile
    const int lane    = tid & 31;
    const int colBase = wave * 16;
    const int m       = lane & 15;
    const int khalf   = (lane >> 4) * 2;   // 0 or 2

    v8f acc = {};
#pragma unroll
    for (int k = 0; k < HID; k += 4) {
        v2f a, b;
        a.x = Atile[m * HID + k + khalf + 0];
        a.y = Atile[m * HID + k + khalf + 1];
        b.x = W[(k + khalf + 0) * HID + colBase + m];
        b.y = W[(k + khalf + 1) * HID + colBase + m];
        acc = __builtin_amdgcn_wmma_f32_16x16x4_f32(false, a, false, b,
                                                    (short)0, acc, false, false);
    }

    const int rowOff = (lane >> 4) * 8;
    const int col    = colBase + m;
#pragma unroll
    for (int r = 0; r < 8; ++r) {
        int row = rowBase + rowOff + r;
        if (row < n) OUT[(long)row * HID + col] = acc[r];
    }
}

// ---------------------------------------------------------------------------
// Edge scatter: agg[dst] += hw[src] * isd[src]*isd[dst]
// thread = (edge, 4 channels); one wave covers exactly one edge row:
// 32 lanes x 16B = 512B coalesced gather + coalesced atomic burst.
// ---------------------------------------------------------------------------
__global__ __launch_bounds__(256) void scatter_kernel(const float* __restrict__ hw,
                                                      const int* __restrict__ src,
                                                      const int* __restrict__ dst,
                                                      const float* __restrict__ isd,
                                                      float* __restrict__ agg) {
    long idx = (long)blockIdx.x * blockDim.x + threadIdx.x;   // over E*32
    if (idx >= (long)N_EDGES * (HID / 4)) return;
    int e  = (int)(idx >> 5);
    int c4 = (int)(idx & 31) * 4;
    int s = src[e], d = dst[e];
    float norm = isd[s] * isd[d];
    float4 m = *(const float4*)&hw[(long)s * HID + c4];
    float* a = &agg[(long)d * HID + c4];
    atomic_add_f32(a + 0, m.x * norm);
    atomic_add_f32(a + 1, m.y * norm);
    atomic_add_f32(a + 2, m.z * norm);
    atomic_add_f32(a + 3, m.w * norm);
}

// ---------------------------------------------------------------------------
// Combine: h_new = agg + hw * isd^2 + b   (in-place into agg), float4
// ---------------------------------------------------------------------------
__global__ void combine_kernel(const float* __restrict__ hw, const float* __restrict__ isd,
                               const float* __restrict__ b, float* __restrict__ agg, int n) {
    int idx = blockIdx.x * blockDim.x + threadIdx.x;   // over n*32
    if (idx >= n * (HID / 4)) return;
    int i  = idx >> 5;
    int c4 = (idx & 31) * 4;
    float w = isd[i];
    float w2 = w * w;
    float4 av = *(const float4*)&agg[(long)i * HID + c4];
    float4 hv = *(const float4*)&hw[(long)i * HID + c4];
    float4 bv = *(const float4*)&b[c4];
    av.x += hv.x * w2 + bv.x; av.y += hv.y * w2 + bv.y;
    av.z += hv.z * w2 + bv.z; av.w += hv.w * w2 + bv.w;
    *(float4*)&agg[(long)i * HID + c4] = av;
}

// ---------------------------------------------------------------------------
// BatchNorm stats: per-channel sum / sumsq (thread t == channel t, coalesced)
// ---------------------------------------------------------------------------
__global__ __launch_bounds__(128) void bn_stats_kernel(const float* __restrict__ h,
                                                       float* __restrict__ sums, int n) {
    int c = threadIdx.x;
    float s = 0.0f, s2 = 0.0f;
    for (int i = blockIdx.x; i < n; i += gridDim.x) {
        float v = h[(long)i * HID + c];
        s += v;
        s2 += v * v;
    }
    atomic_add_f32(&sums[c], s);
    atomic_add_f32(&sums[HID + c], s2);
}

__global__ __launch_bounds__(128) void bn_finalize_kernel(const float* __restrict__ sums,
                                                          float* __restrict__ stats, int n) {
    int c = threadIdx.x;
    float inv_n = 1.0f / (float)n;
    float mu  = sums[c] * inv_n;
    float var = sums[HID + c] * inv_n - mu * mu;
    stats[c]       = mu;
    stats[HID + c] = rsqrtf(var + BN_EPS);
}

// encoder: h = bn(h)  (relu already applied), float4
__global__ void bn_apply_kernel(float* __restrict__ h, const float* __restrict__ stats,
                                const float* __restrict__ g, const float* __restrict__ b, int n) {
    int idx = blockIdx.x * blockDim.x + threadIdx.x;
    if (idx >= n * (HID / 4)) return;
    int i  = idx >> 5;
    int c4 = (idx & 31) * 4;
    float4 hv = *(const float4*)&h[(long)i * HID + c4];
    float4 mu = *(const float4*)&stats[c4];
    float4 rs = *(const float4*)&stats[HID + c4];
    float4 gv = *(const float4*)&g[c4];
    float4 bv = *(const float4*)&b[c4];
    hv.x = (hv.x - mu.x) * rs.x * gv.x + bv.x;
    hv.y = (hv.y - mu.y) * rs.y * gv.y + bv.y;
    hv.z = (hv.z - mu.z) * rs.z * gv.z + bv.z;
    hv.w = (hv.w - mu.w) * rs.w * gv.w + bv.w;
    *(float4*)&h[(long)i * HID + c4] = hv;
}

// layer: h += relu(bn(h_new)), float4
__global__ void bn_relu_residual_kernel(const float* __restrict__ hnew,
                                        const float* __restrict__ stats,
                                        const float* __restrict__ g, const float* __restrict__ b,
                                        float* __restrict__ h, int n) {
    int idx = blockIdx.x * blockDim.x + threadIdx.x;
    if (idx >= n * (HID / 4)) return;
    int i  = idx >> 5;
    int c4 = (idx & 31) * 4;
    float4 nv = *(const float4*)&hnew[(long)i * HID + c4];
    float4 mu = *(const float4*)&stats[c4];
    float4 rs = *(const float4*)&stats[HID + c4];
    float4 gv = *(const float4*)&g[c4];
    float4 bv = *(const float4*)&b[c4];
    float4 hv = *(const float4*)&h[(long)i * HID + c4];
    hv.x += fmaxf((nv.x - mu.x) * rs.x * gv.x + bv.x, 0.f);
    hv.y += fmaxf((nv.y - mu.y) * rs.y * gv.y + bv.y, 0.f);
    hv.z += fmaxf((nv.z - mu.z) * rs.z * gv.z + bv.z, 0.f);
    hv.w += fmaxf((nv.w - mu.w) * rs.w * gv.w + bv.w, 0.f);
    *(float4*)&h[(long)i * HID + c4] = hv;
}

// ---------------------------------------------------------------------------
// Fused decoder MLP: 128 -> 64 -> 32 -> 1, one node per 64-thread block
// ---------------------------------------------------------------------------
__global__ __launch_bounds__(64) void decoder_kernel(const float* __restrict__ h,
                                                     const float* __restrict__ w1, const float* __restrict__ b1,
                                                     const float* __restrict__ w2, const float* __restrict__ b2,
                                                     const float* __restrict__ w3, const float* __restrict__ b3,
                                                     float* __restrict__ out, int n) {
    __shared__ float hrow[HID];
    __shared__ float d1[64];
    __shared__ float d2[32];
    int i = blockIdx.x;
    if (i >= n) return;
    int t = threadIdx.x;
    hrow[t]      = h[(long)i * HID + t];
    hrow[t + 64] = h[(long)i * HID + t + 64];
    __syncthreads();

    float acc = b1[t];
#pragma unroll 8
    for (int k = 0; k < HID; ++k) acc += hrow[k] * w1[k * 64 + t];
    d1[t] = fmaxf(acc, 0.0f);
    __syncthreads();

    if (t < 32) {
        float a2 = b2[t];
#pragma unroll 8
        for (int k = 0; k < 64; ++k) a2 += d1[k] * w2[k * 32 + t];
        d2[t] = fmaxf(a2, 0.0f);
    }
    __syncthreads();

    if (t == 0) {
        float o = b3[0];
#pragma unroll
        for (int k = 0; k < 32; ++k) o += d2[k] * w3[k];
        out[i] = o;
    }
}

// ---------------------------------------------------------------------------
// Host orchestration
// ---------------------------------------------------------------------------
extern "C" void kernel_launch(void* const* d_in, const int* in_sizes, int n_in,
                              void* d_out, int out_size, void* d_ws, size_t ws_size,
                              hipStream_t stream) {
    (void)in_sizes; (void)n_in; (void)out_size; (void)ws_size;

    const float* x       = (const float*)d_in[0];
    const int*   ei      = (const int*)d_in[1];      // [2, E]
    const float* enc_w   = (const float*)d_in[2];
    const float* enc_b   = (const float*)d_in[3];
    const float* enc_bng = (const float*)d_in[4];
    const float* enc_bnb = (const float*)d_in[5];
    const float* conv_w  = (const float*)d_in[6];    // [L,128,128]
    const float* conv_b  = (const float*)d_in[7];    // [L,128]
    const float* bn_g    = (const float*)d_in[8];
    const float* bn_b    = (const float*)d_in[9];
    const float* dec_w1  = (const float*)d_in[10];
    const float* dec_b1  = (const float*)d_in[11];
    const float* dec_w2  = (const float*)d_in[12];
    const float* dec_b2  = (const float*)d_in[13];
    const float* dec_w3  = (const float*)d_in[14];
    const float* dec_b3  = (const float*)d_in[15];
    float* out = (float*)d_out;

    const int* src = ei;
    const int* dst = ei + N_EDGES;

    // workspace layout (all segment offsets are 16B-aligned)
    float* ws   = (float*)d_ws;
    float* deg  = ws;                               // N
    float* isd  = deg + N_NODES;                    // N
    float* h    = isd + N_NODES;                    // N*HID
    float* hw   = h   + (size_t)N_NODES * HID;      // N*HID
    float* agg  = hw  + (size_t)N_NODES * HID;      // N*HID
    float* sums = agg + (size_t)N_NODES * HID;      // 2*HID
    float* stats= sums + 2 * HID;                   // 2*HID

    const int elemN    = N_NODES * HID;
    const int vec4N    = N_NODES * (HID / 4);
    const int vec4Blk  = (vec4N + 255) / 256;
    const int gemmBlk  = (N_NODES + 15) / 16;
    const long scatterTotal = (long)N_EDGES * (HID / 4);
    const int scatterBlk = (int)((scatterTotal + 255) / 256);

    // ---- degree / normalization ----
    hipMemsetAsync(deg, 0, N_NODES * sizeof(float), stream);
    degree_kernel<<<(N_EDGES + 255) / 256, 256, 0, stream>>>(dst, deg, N_EDGES);
    inv_sqrt_deg_kernel<<<(N_NODES + 255) / 256, 256, 0, stream>>>(deg, isd, N_NODES);

    // ---- encoder: relu(linear) then BN ----
    encoder_kernel<<<vec4Blk, 256, 0, stream>>>(x, enc_w, enc_b, h, N_NODES);
    hipMemsetAsync(sums, 0, 2 * HID * sizeof(float), stream);
    bn_stats_kernel<<<512, 128, 0, stream>>>(h, sums, N_NODES);
    bn_finalize_kernel<<<1, 128, 0, stream>>>(sums, stats, N_NODES);
    bn_apply_kernel<<<vec4Blk, 256, 0, stream>>>(h, stats, enc_bng, enc_bnb, N_NODES);

    // ---- GCN layers ----
    for (int l = 0; l < N_LAYERS; ++l) {
        const float* Wl = conv_w + (size_t)l * HID * HID;
        const float* bl = conv_b + (size_t)l * HID;
        const float* gl = bn_g   + (size_t)l * HID;
        const float* bbl= bn_b   + (size_t)l * HID;

        gemm_wmma_f32<<<gemmBlk, 256, 0, stream>>>(h, Wl, hw, N_NODES);
        hipMemsetAsync(agg, 0, (size_t)elemN * sizeof(float), stream);
        scatter_kernel<<<scatterBlk, 256, 0, stream>>>(hw, src, dst, isd, agg);
        combine_kernel<<<vec4Blk, 256, 0, stream>>>(hw, isd, bl, agg, N_NODES);

        hipMemsetAsync(sums, 0, 2 * HID * sizeof(float), stream);
        bn_stats_kernel<<<512, 128, 0, stream>>>(agg, sums, N_NODES);
        bn_finalize_kernel<<<1, 128, 0, stream>>>(sums, stats, N_NODES);
        bn_relu_residual_kernel<<<vec4Blk, 256, 0, stream>>>(agg, stats, gl, bbl, h, N_NODES);
    }

    // ---- decoder MLP ----
    decoder_kernel<<<N_NODES, 64, 0, stream>>>(h, dec_w1, dec_b1, dec_w2, dec_b2,
                                               dec_w3, dec_b3, out, N_NODES);
}